// SEAttentionGNN_71614284693670
// MI455X (gfx1250) — compile-verified
//
#include <hip/hip_runtime.h>
#include <math.h>

typedef __attribute__((ext_vector_type(2))) float v2f;
typedef __attribute__((ext_vector_type(8))) float v8f;

#define BB 32
#define CC 128
#define HH 96
#define WW 96
#define NN (HH * WW)   /* 9216 */
#define CR 8           /* C / r */
#define NT 64          /* n-tile per workgroup */
#define ZST 132        /* LDS row stride (floats), even + non-multiple-of-64 */

// ---------------------------------------------------------------------------
// Kernel 1a: per-(b,c) channel mean over H*W (one block per channel, contiguous)
// ---------------------------------------------------------------------------
__global__ __launch_bounds__(256)
void se_mean_kernel(const float* __restrict__ x, float* __restrict__ y0) {
    const int bc = blockIdx.x;                   // 0 .. B*C-1
    const float* xc = x + (size_t)bc * NN;
    float s = 0.0f;
    for (int i = threadIdx.x; i < NN; i += 256) s += xc[i];
    __shared__ float red[256];
    red[threadIdx.x] = s;
    __syncthreads();
    for (int off = 128; off > 0; off >>= 1) {
        if (threadIdx.x < off) red[threadIdx.x] += red[threadIdx.x + off];
        __syncthreads();
    }
    if (threadIdx.x == 0) y0[bc] = red[0] * (1.0f / (float)NN);
}

// ---------------------------------------------------------------------------
// Kernel 1b: SE MLP  y = sigmoid(relu(y0 @ W1^T) @ W2^T)   (one block per batch)
// ---------------------------------------------------------------------------
__global__ __launch_bounds__(CC)
void se_mlp_kernel(const float* __restrict__ y0,
                   const float* __restrict__ w_fc1,   // (CR, C) row-major
                   const float* __restrict__ w_fc2,   // (C, CR) row-major
                   float* __restrict__ y) {
    const int b = blockIdx.x;
    const int c = threadIdx.x;                   // 0..127
    __shared__ float ym[CC];
    __shared__ float hb[CR];
    ym[c] = y0[b * CC + c];
    __syncthreads();
    if (c < CR) {
        float h = 0.0f;
        for (int k = 0; k < CC; ++k) h += w_fc1[c * CC + k] * ym[k];
        hb[c] = fmaxf(h, 0.0f);
    }
    __syncthreads();
    float o = 0.0f;
    #pragma unroll
    for (int j = 0; j < CR; ++j) o += w_fc2[c * CR + j] * hb[j];
    y[b * CC + c] = 1.0f / (1.0f + __expf(-o));
}

// ---------------------------------------------------------------------------
// Kernel 2: fused   out^T = W^T * (S * (x .* gate))^T  + bias
//   stage 1: z[n_local][c_in] = dinv(n) * ( dinv(n)*xg(n)
//                                         + dinv(n-1)*xg(n-1)  (if j>0)
//                                         + dinv(n-W)*xg(n-W)  (if i>0) )
//   stage 2: per-wave V_WMMA_F32_16X16X4_F32 over K = c_in = 128
//   stage 3: store directly into (b, c_out, h, w)
// ---------------------------------------------------------------------------
__device__ __forceinline__ float dinv_ij(int i, int j) {
    // in-degree = self + (up edge exists) + (left edge exists)
    const float d = 1.0f + (i > 0 ? 1.0f : 0.0f) + (j > 0 ? 1.0f : 0.0f);
    return rsqrtf(d);
}

__global__ __launch_bounds__(256)
void gcn_wmma_kernel(const float* __restrict__ x,      // (B, C, N)
                     const float* __restrict__ ygate,  // (B, C)
                     const float* __restrict__ gcn_w,  // (C_in, C_out) row-major
                     const float* __restrict__ gcn_b,  // (C)
                     float* __restrict__ out) {        // (B, C, N)
    __shared__ __align__(16) float zt[NT * ZST];

    const int b   = blockIdx.y;
    const int n0  = blockIdx.x * NT;
    const int tid = threadIdx.x;

    // ---- stage 1: build gated+aggregated z tile in LDS, layout [n_local][c] ----
    {
        const int nl = tid & 63;       // n_local within tile
        const int cg = tid >> 6;       // channel group 0..3
        const int n  = n0 + nl;
        const int i  = n / WW;
        const int j  = n - i * WW;
        const float s  = dinv_ij(i, j);
        const float sl = (j > 0) ? dinv_ij(i, j - 1) : 0.0f;
        const float su = (i > 0) ? dinv_ij(i - 1, j) : 0.0f;
        for (int c = cg; c < CC; c += 4) {
            const float* xc = x + ((size_t)b * CC + c) * NN;
            float acc = s * xc[n];
            if (j > 0) acc += sl * xc[n - 1];
            if (i > 0) acc += su * xc[n - WW];
            zt[nl * ZST + c] = ygate[b * CC + c] * s * acc;
        }
    }
    __syncthreads();

    // ---- stage 2: WMMA. Each wave: one 16-wide c_out block x four 16-wide n blocks ----
    const int lane  = tid & 31;
    const int wave  = tid >> 5;           // 0..7
    const int cbase = wave * 16;          // c_out block base
    const int lm    = lane & 15;
    const int h2    = (lane >> 4) << 1;   // 0 (lanes 0-15) or 2 (lanes 16-31)

    v8f acc0 = {}, acc1 = {}, acc2 = {}, acc3 = {};

    #pragma unroll 4
    for (int k0 = 0; k0 < CC; k0 += 4) {
        // A = W^T tile (16 c_out x 4 c_in): A[m][k] = gcn_w[k*C + cbase + m]
        // lane layout: VGPR0 = K (k0 | k0+2), VGPR1 = K+1  -> coalesced b32 loads
        v2f a;
        a.x = gcn_w[(k0 + h2    ) * CC + cbase + lm];
        a.y = gcn_w[(k0 + h2 + 1) * CC + cbase + lm];
        // B = z^T tile (4 c_in x 16 n): one aligned 8-byte LDS load per subtile
        const v2f b0 = *(const v2f*)&zt[( 0 + lm) * ZST + k0 + h2];
        const v2f b1 = *(const v2f*)&zt[(16 + lm) * ZST + k0 + h2];
        const v2f b2 = *(const v2f*)&zt[(32 + lm) * ZST + k0 + h2];
        const v2f b3 = *(const v2f*)&zt[(48 + lm) * ZST + k0 + h2];
        acc0 = __builtin_amdgcn_wmma_f32_16x16x4_f32(false, a, false, b0, (short)0, acc0, false, false);
        acc1 = __builtin_amdgcn_wmma_f32_16x16x4_f32(false, a, false, b1, (short)0, acc1, false, false);
        acc2 = __builtin_amdgcn_wmma_f32_16x16x4_f32(false, a, false, b2, (short)0, acc2, false, false);
        acc3 = __builtin_amdgcn_wmma_f32_16x16x4_f32(false, a, false, b3, (short)0, acc3, false, false);
    }

    // ---- stage 3: bias + store straight into (b, c_out, n) ----
    const int hi8 = (lane >> 4) << 3;     // 0 or 8 (C-tile half-wave row offset)
    #pragma unroll
    for (int v = 0; v < 8; ++v) {
        const int c_out = cbase + v + hi8;
        const float bias = gcn_b[c_out];
        float* orow = out + ((size_t)b * CC + c_out) * NN + n0 + lm;
        orow[ 0] = acc0[v] + bias;
        orow[16] = acc1[v] + bias;
        orow[32] = acc2[v] + bias;
        orow[48] = acc3[v] + bias;
    }
}

// ---------------------------------------------------------------------------
extern "C" void kernel_launch(void* const* d_in, const int* in_sizes, int n_in,
                              void* d_out, int out_size, void* d_ws, size_t ws_size,
                              hipStream_t stream) {
    (void)in_sizes; (void)n_in; (void)out_size; (void)ws_size;
    const float* x     = (const float*)d_in[0];
    const float* w_fc1 = (const float*)d_in[1];
    const float* w_fc2 = (const float*)d_in[2];
    const float* gcn_w = (const float*)d_in[3];
    const float* gcn_b = (const float*)d_in[4];
    float* out = (float*)d_out;

    float* y0 = (float*)d_ws;            // B*C floats
    float* y  = y0 + BB * CC;            // B*C floats

    se_mean_kernel<<<BB * CC, 256, 0, stream>>>(x, y0);
    se_mlp_kernel<<<BB, CC, 0, stream>>>(y0, w_fc1, w_fc2, y);

    dim3 grid(NN / NT, BB);
    gcn_wmma_kernel<<<grid, 256, 0, stream>>>(x, y, gcn_w, gcn_b, out);
}